// GroupPoints_euclidean_density_60705067761860
// MI455X (gfx1250) — compile-verified
//
#include <hip/hip_runtime.h>
#include <hip/hip_bf16.h>
#include <math.h>

typedef __attribute__((ext_vector_type(2))) float v2f;
typedef __attribute__((ext_vector_type(8))) float v8f;

#define KNN     32
#define TILE_T  32              // targets per block (full wave of selectors)
#define CHUNK   64              // sources per chunk
#define NTHR    256             // 8 waves: waves 0-3 -> targets 0-15, waves 4-7 -> targets 16-31

__global__ __launch_bounds__(NTHR)
void group_points_knn_kernel(const float* __restrict__ source,
                             const float* __restrict__ target,
                             const float* __restrict__ source_density,
                             float* __restrict__ out,
                             int B, int Ns, int Nt)
{
    __shared__ float sTx[TILE_T], sTy[TILE_T], sTz[TILE_T], sR0[TILE_T];
    __shared__ float sDist[TILE_T][CHUNK];
    __shared__ float sTopD[TILE_T][KNN];
    __shared__ int   sTopI[TILE_T][KNN];

    const int tid    = threadIdx.x;
    const int wave   = tid >> 5;
    const int lane   = tid & 31;
    const int half   = lane >> 4;          // 0: lanes 0-15, 1: lanes 16-31
    const int ln     = lane & 15;
    const int tgroup = wave >> 2;          // which 16-target group this wave computes
    const int wsub   = wave & 3;           // source sub-tile within the chunk

    const int blk = blockIdx.x;            // 0 .. B*Nt/32 - 1
    const int gT0 = blk * TILE_T;          // flattened (b, t) base
    const int b   = gT0 / Nt;
    const int t0  = gT0 - b * Nt;          // target base inside batch

    // ---- load target tile, precompute |t|^2, init top-k sentinels ----
    if (tid < TILE_T) {
        const float* tp = target + ((size_t)b * Nt + (t0 + tid)) * 3;
        float x = tp[0], y = tp[1], z = tp[2];
        sTx[tid] = x; sTy[tid] = y; sTz[tid] = z;
        sR0[tid] = x * x + y * y + z * z;
    }
    {
        // fill 32x32 lists with +BIG so the scan loop needs no warm-up branch
        for (int i = tid; i < TILE_T * KNN; i += NTHR) {
            ((float*)sTopD)[i] = 3.0e38f;
            ((int*)sTopI)[i]   = -1;
        }
    }
    __syncthreads();

    // ---- A operand (16x4 f32): lanes 0-15 carry K0,K1 = (x,y); lanes 16-31 carry K2,K3 = (z,0)
    const int mBase = tgroup * 16;
    v2f amat;
    if (half == 0) { amat[0] = sTx[mBase + ln]; amat[1] = sTy[mBase + ln]; }
    else           { amat[0] = sTz[mBase + ln]; amat[1] = 0.0f;            }

    // |t|^2 for the C/D rows this lane owns: row = mBase + v + 8*half
    float r0v[8];
#pragma unroll
    for (int v = 0; v < 8; ++v) r0v[v] = sR0[mBase + v + 8 * half];

    float worst = 3.0e38f;                 // per-selector running 32nd-best bound

    const int nChunks = Ns / CHUNK;        // 2048/64 = 32
    for (int c = 0; c < nChunks; ++c) {
        const int sBase = c * CHUNK;
        const int sIdx  = sBase + wsub * 16 + ln;

        const float* sp = source + ((size_t)b * Ns + sIdx) * 3;
        float sx = sp[0], sy = sp[1], sz = sp[2];
        float r1 = sx * sx + sy * sy + sz * sz;

        // B operand (4x16 f32), N = ln
        v2f bmat;
        if (half == 0) { bmat[0] = sx; bmat[1] = sy; }
        else           { bmat[0] = sz; bmat[1] = 0.0f; }

        v8f acc = {};
        acc = __builtin_amdgcn_wmma_f32_16x16x4_f32(
            /*neg_a=*/false, amat, /*neg_b=*/false, bmat,
            /*c_mod=*/(short)0, acc, /*reuse_a=*/false, /*reuse_b=*/false);

        // sq_dist = |t|^2 + |s|^2 - 2*dot ; scatter tile rows into LDS
#pragma unroll
        for (int v = 0; v < 8; ++v) {
            float d = r0v[v] + r1 - 2.0f * acc[v];
            sDist[mBase + v + 8 * half][wsub * 16 + ln] = d;
        }
        __syncthreads();

        // ---- wave-wide stable top-32 scan: lane == target row ----
        if (tid < TILE_T) {
            for (int j = 0; j < CHUNK; ++j) {
                float d = sDist[tid][j];
                if (d < worst) {           // rare after warm-up
                    int si = sBase + j;
                    int p = KNN - 1;
                    while (p > 0 && sTopD[tid][p - 1] > d) {
                        sTopD[tid][p] = sTopD[tid][p - 1];
                        sTopI[tid][p] = sTopI[tid][p - 1];
                        --p;
                    }
                    sTopD[tid][p] = d; sTopI[tid][p] = si;
                    worst = sTopD[tid][KNN - 1];
                }
            }
        }
        __syncthreads();
    }

    // ---- emit outputs (concatenated flat, reference return order) ----
    if (tid < TILE_T) {
        const int t = t0 + tid;

        const size_t sz1 = (size_t)B * Nt * KNN * 3;   // patches
        const size_t sz2 = (size_t)B * Nt * KNN * 2;   // patches_idx
        const size_t sz3 = (size_t)B * Nt;             // patches_size
        const size_t sz4 = (size_t)B;                  // rad (B,1,1)
        const size_t sz5 = (size_t)B * Nt * KNN;       // patches_dist

        float* o1 = out;
        float* o2 = o1 + sz1;
        float* o3 = o2 + sz2;
        float* o4 = o3 + sz3;
        float* o5 = o4 + sz4;
        float* o6 = o5 + sz5;

        const float rad  = 0.08f;
        const float rad2 = rad * rad;
        const float inv  = 1.0f / (rad + 1e-6f);

        const float tx = sTx[tid] * inv;
        const float ty = sTy[tid] * inv;
        const float tz = sTz[tid] * inv;

        int   nvalid  = 0;
        float densSum = 0.0f;
        const size_t base = ((size_t)b * Nt + t) * KNN;

        for (int k = 0; k < KNN; ++k) {
            float d  = sTopD[tid][k];
            int   si = sTopI[tid][k];
            bool valid = (rad2 >= d);
            float px = 0.0f, py = 0.0f, pz = 0.0f, dens = 0.0f;
            if (valid) {
                const float* sp = source + ((size_t)b * Ns + si) * 3;
                px = sp[0] * inv; py = sp[1] * inv; pz = sp[2] * inv;
                dens = source_density[(size_t)b * Ns + si];
                ++nvalid;
            }
            o1[(base + k) * 3 + 0] = px - tx;
            o1[(base + k) * 3 + 1] = py - ty;
            o1[(base + k) * 3 + 2] = pz - tz;
            o2[(base + k) * 2 + 0] = (float)b;
            o2[(base + k) * 2 + 1] = (float)(valid ? si : -1);
            float dc = d > 1e-9f ? d : 1e-9f;
            o5[base + k] = sqrtf(dc) * inv;
            densSum += dens;
        }
        o3[(size_t)b * Nt + t] = (float)nvalid;
        o6[(size_t)b * Nt + t] = densSum / (float)(KNN - nvalid);

        if (blk == 0 && tid == 0) {
            for (int i = 0; i < B; ++i) o4[i] = rad;
        }
    }
}

extern "C" void kernel_launch(void* const* d_in, const int* in_sizes, int n_in,
                              void* d_out, int out_size, void* d_ws, size_t ws_size,
                              hipStream_t stream)
{
    const float* source         = (const float*)d_in[0];
    const float* target         = (const float*)d_in[1];
    const float* source_density = (const float*)d_in[2];
    // d_in[3] (target_density) feeds only the dead _sq_dist_sel value: unused.
    float* out = (float*)d_out;

    const int B  = 2;
    const int Ns = in_sizes[0] / (B * 3);   // 2048
    const int Nt = in_sizes[1] / (B * 3);   // 32768

    const int blocks = (B * Nt) / TILE_T;   // 2048
    hipLaunchKernelGGL(group_points_knn_kernel, dim3(blocks), dim3(NTHR), 0, stream,
                       source, target, source_density, out, B, Ns, Nt);
}